// GAT_MLP_12610023981850
// MI455X (gfx1250) — compile-verified
//
#include <hip/hip_runtime.h>
#include <math.h>

#define NFEATC   64
#define NHIDC    64
#define NHEADSC  8
#define HFEATC   512      /* NHEADS*NHID */
#define MLPH     640      /* NFEAT*10 */
#define LSLOPE   0.2f

typedef __attribute__((ext_vector_type(2))) float v2f;
typedef __attribute__((ext_vector_type(8))) float v8f;

// -------- fp32 WMMA helper: C(16x16) += A(16xK) * B(Kx16) ------------------
// A row-major with leading dim lda, B row-major with leading dim ldb.
// Fragment layouts per CDNA5 ISA 7.12.2 (32-bit A 16x4 / 32-bit C 16x16).
__device__ __forceinline__ v8f wmma_tile_f32(const float* __restrict__ A, int lda,
                                             const float* __restrict__ B, int ldb,
                                             int K, v8f acc) {
  const int lane = threadIdx.x & 31;
  const int half = lane >> 4;   // 0: lanes 0-15, 1: lanes 16-31
  const int idx  = lane & 15;
  for (int k0 = 0; k0 < K; k0 += 4) {
    const int kk = k0 + 2 * half;
    v2f a, b;
    a.x = A[idx * lda + kk];
    a.y = A[idx * lda + kk + 1];
    b.x = B[kk * ldb + idx];
    b.y = B[(kk + 1) * ldb + idx];
    acc = __builtin_amdgcn_wmma_f32_16x16x4_f32(false, a, false, b, (short)0, acc,
                                                false, false);
  }
  return acc;
}

// -------- K1: h1[n, head*64+d] = (x @ W_att[head])[n,d] --------------------
__global__ void k_gemm_heads(const float* __restrict__ x,
                             const float* __restrict__ Watt,
                             float* __restrict__ h1, int ntiles) {
  const int wave  = (blockIdx.x * blockDim.x + threadIdx.x) >> 5;
  const int ntile = wave >> 3;
  const int head  = wave & 7;
  if (ntile >= ntiles) return;
  const int lane = threadIdx.x & 31;
  const int half = lane >> 4, idx = lane & 15;
  const float* A = x + (size_t)ntile * 16 * NFEATC;
  const float* W = Watt + (size_t)head * NFEATC * NHIDC;
  for (int nt = 0; nt < NHIDC / 16; ++nt) {
    v8f acc = {};
    acc = wmma_tile_f32(A, NFEATC, W + nt * 16, NHIDC, NFEATC, acc);
    const int ncol = nt * 16 + idx;
    for (int r = 0; r < 8; ++r) {
      const int m = r + half * 8;
      h1[(size_t)(ntile * 16 + m) * HFEATC + head * NHIDC + ncol] = acc[r];
    }
  }
}

// -------- K2: per-(node,head) attention logits -----------------------------
__global__ void k_alpha1(const float* __restrict__ h1,
                         const float* __restrict__ a_att,
                         float* __restrict__ asrc, float* __restrict__ adst,
                         int n_nodes) {
  const int t = blockIdx.x * blockDim.x + threadIdx.x;
  if (t >= n_nodes * NHEADSC) return;
  const int n = t >> 3, h = t & 7;
  const float* hv = h1 + (size_t)n * HFEATC + h * NHIDC;
  const float* as = a_att + h * 2 * NHIDC;
  const float* ad = as + NHIDC;
  float s = 0.f, d = 0.f;
  for (int k = 0; k < NHIDC; ++k) { float v = hv[k]; s += v * as[k]; d += v * ad[k]; }
  asrc[t] = s; adst[t] = d;
}

// -------- K3: layer-1 edge scatter (one wave32 per edge) -------------------
__global__ void k_edge1(const int* __restrict__ row, const int* __restrict__ col,
                        const float* __restrict__ asrc, const float* __restrict__ adst,
                        const float* __restrict__ h1,
                        float* __restrict__ hp1, float* __restrict__ rowsum, int E) {
  const int wave = (blockIdx.x * blockDim.x + threadIdx.x) >> 5;
  if (wave >= E) return;
  const int lane = threadIdx.x & 31;
  const int r = row[wave], c = col[wave];
  float ee = 0.f;
  if (lane < NHEADSC) {
    const float s  = asrc[r * NHEADSC + lane] + adst[c * NHEADSC + lane];
    const float lr = s > 0.f ? s : LSLOPE * s;
    ee = __expf(-lr);
    atomicAdd(&rowsum[r * NHEADSC + lane], ee);
  }
  const float* hc = h1 + (size_t)c * HFEATC;
  float* hr       = hp1 + (size_t)r * HFEATC;
  for (int i = 0; i < HFEATC / 32; ++i) {           // head = i>>1 per iteration
    const float e = __shfl(ee, i >> 1, 32);
    const int f = i * 32 + lane;
    atomicAdd(&hr[f], e * hc[f]);
  }
}

// -------- K4: normalize + ELU (in place: hp1 -> xh) ------------------------
__global__ void k_norm1(float* __restrict__ hp, const float* __restrict__ rowsum,
                        int total) {
  const int t = blockIdx.x * blockDim.x + threadIdx.x;
  if (t >= total) return;
  const int n = t / HFEATC, f = t % HFEATC, h = f >> 6;
  const float v = hp[t] / rowsum[n * NHEADSC + h];
  hp[t] = v > 0.f ? v : (__expf(v) - 1.f);
}

// -------- K5: h2 = xh @ W_out  ([N,512]@[512,64]) --------------------------
__global__ void k_gemm2(const float* __restrict__ xh,
                        const float* __restrict__ Wout,
                        float* __restrict__ h2, int ntiles) {
  const int wave  = (blockIdx.x * blockDim.x + threadIdx.x) >> 5;
  const int ntile = wave >> 2;
  const int nt    = wave & 3;
  if (ntile >= ntiles) return;
  const int lane = threadIdx.x & 31;
  const int half = lane >> 4, idx = lane & 15;
  const float* A = xh + (size_t)ntile * 16 * HFEATC;
  v8f acc = {};
  acc = wmma_tile_f32(A, HFEATC, Wout + nt * 16, NHIDC, HFEATC, acc);
  const int ncol = nt * 16 + idx;
  for (int r = 0; r < 8; ++r) {
    const int m = r + half * 8;
    h2[(size_t)(ntile * 16 + m) * NHIDC + ncol] = acc[r];
  }
}

// -------- K6: layer-2 logits (thread per node) -----------------------------
__global__ void k_alpha2(const float* __restrict__ h2,
                         const float* __restrict__ a_out,
                         float* __restrict__ asrc, float* __restrict__ adst,
                         int n_nodes) {
  const int n = blockIdx.x * blockDim.x + threadIdx.x;
  if (n >= n_nodes) return;
  const float* hv = h2 + (size_t)n * NHIDC;
  float s = 0.f, d = 0.f;
  for (int k = 0; k < NHIDC; ++k) { float v = hv[k]; s += v * a_out[k]; d += v * a_out[NHIDC + k]; }
  asrc[n] = s; adst[n] = d;
}

// -------- K7: layer-2 edge scatter -----------------------------------------
__global__ void k_edge2(const int* __restrict__ row, const int* __restrict__ col,
                        const float* __restrict__ asrc, const float* __restrict__ adst,
                        const float* __restrict__ h2,
                        float* __restrict__ hp2, float* __restrict__ rowsum, int E) {
  const int wave = (blockIdx.x * blockDim.x + threadIdx.x) >> 5;
  if (wave >= E) return;
  const int lane = threadIdx.x & 31;
  const int r = row[wave], c = col[wave];
  float ee = 0.f;
  if (lane == 0) {
    const float s  = asrc[r] + adst[c];
    const float lr = s > 0.f ? s : LSLOPE * s;
    ee = __expf(-lr);
    atomicAdd(&rowsum[r], ee);
  }
  const float e = __shfl(ee, 0, 32);
  const float* hc = h2 + (size_t)c * NHIDC;
  float* hr       = hp2 + (size_t)r * NHIDC;
  atomicAdd(&hr[lane],      e * hc[lane]);
  atomicAdd(&hr[lane + 32], e * hc[lane + 32]);
}

// -------- K8: xo = elu(hp2 / rowsum2) --------------------------------------
__global__ void k_norm2(const float* __restrict__ hp, const float* __restrict__ rowsum,
                        float* __restrict__ xo, int total) {
  const int t = blockIdx.x * blockDim.x + threadIdx.x;
  if (t >= total) return;
  const int n = t / NHIDC;
  const float v = hp[t] / rowsum[n];
  xo[t] = v > 0.f ? v : (__expf(v) - 1.f);
}

// -------- K9: fused MLP head (one wave per 16-node tile) -------------------
__global__ void __launch_bounds__(32)
k_mlp(const float* __restrict__ xo,
      const float* __restrict__ W1, const float* __restrict__ b1,
      const float* __restrict__ W2, const float* __restrict__ b2,
      const float* __restrict__ W3, const float* __restrict__ b3,
      float* __restrict__ out, int ntiles) {
  __shared__ float z1s[16 * MLPH];   // 40 KB
  __shared__ float z2s[16 * 10];
  const int tile = blockIdx.x;
  if (tile >= ntiles) return;
  const int lane = threadIdx.x & 31;
  const int half = lane >> 4, idx = lane & 15;
  const float* A = xo + (size_t)tile * 16 * NFEATC;

  // z1 = relu(xo @ W1 + b1): 40 tiles of 16x16, K=64, WMMA fp32
  for (int nt = 0; nt < MLPH / 16; ++nt) {
    v8f acc = {};
    acc = wmma_tile_f32(A, NFEATC, W1 + nt * 16, MLPH, NFEATC, acc);
    const int ncol = nt * 16 + idx;
    const float bias = b1[ncol];
    for (int r = 0; r < 8; ++r) {
      const int m = r + half * 8;
      float v = acc[r] + bias;
      z1s[m * MLPH + ncol] = v > 0.f ? v : 0.f;
    }
  }
  __syncthreads();

  // z2 = relu(z1 @ W2 + b2): 16 nodes x 10 outs = 160 dots of length 640
  for (int p = lane; p < 160; p += 32) {
    const int n = p / 10, o = p % 10;
    float s = b2[o];
    const float* zrow = &z1s[n * MLPH];
    for (int j = 0; j < MLPH; ++j) s += zrow[j] * W2[j * 10 + o];
    z2s[n * 10 + o] = s > 0.f ? s : 0.f;
  }
  __syncthreads();

  // out = sigmoid(z2 @ W3 + b3)
  if (lane < 16) {
    float s = b3[0];
    for (int o = 0; o < 10; ++o) s += z2s[lane * 10 + o] * W3[o];
    out[tile * 16 + lane] = 1.f / (1.f + __expf(-s));
  }
}

// ---------------------------------------------------------------------------
extern "C" void kernel_launch(void* const* d_in, const int* in_sizes, int n_in,
                              void* d_out, int out_size, void* d_ws, size_t ws_size,
                              hipStream_t stream) {
  (void)n_in; (void)out_size; (void)ws_size;
  const float* x     = (const float*)d_in[0];
  const int*   eidx  = (const int*)  d_in[1];
  const float* Watt  = (const float*)d_in[2];
  const float* a_att = (const float*)d_in[3];
  const float* Wout  = (const float*)d_in[4];
  const float* a_out = (const float*)d_in[5];
  const float* W1    = (const float*)d_in[6];
  const float* b1    = (const float*)d_in[7];
  const float* W2    = (const float*)d_in[8];
  const float* b2    = (const float*)d_in[9];
  const float* W3    = (const float*)d_in[10];
  const float* b3    = (const float*)d_in[11];
  float* out = (float*)d_out;

  const int N = in_sizes[0] / NFEATC;       // 100000
  const int E = in_sizes[1] / 2;            // 1,600,000
  const int ntiles = N / 16;                // 6250 (N % 16 == 0)
  const int* row = eidx;
  const int* col = eidx + E;

  // ---- workspace layout (floats); layer-2 buffers alias dead h1 region ----
  float* ws = (float*)d_ws;
  const size_t NS   = (size_t)N * HFEATC;   // 512N
  float* h1   = ws;                         // [N,512]  (region A)
  float* hp1  = ws + NS;                    // [N,512]  -> xh after norm
  float* asc1 = ws + 2 * NS;                // [N,8]
  float* ads1 = asc1 + (size_t)N * 8;       // [N,8]
  float* rs1  = ads1 + (size_t)N * 8;       // [N,8]
  // layer-2 buffers live inside region A (h1 dead after k_edge1):
  float* h2   = h1;                         // [N,64]
  float* hp2  = h1 + (size_t)N * 64;        // [N,64]
  float* xov  = h1 + (size_t)N * 128;       // [N,64]
  float* asc2 = h1 + (size_t)N * 192;       // [N]
  float* ads2 = asc2 + N;                   // [N]
  float* rs2  = ads2 + N;                   // [N]

  // ---- layer 1 ----
  hipMemsetAsync(hp1, 0, NS * sizeof(float), stream);
  hipMemsetAsync(rs1, 0, (size_t)N * 8 * sizeof(float), stream);

  k_gemm_heads<<<ntiles, 256, 0, stream>>>(x, Watt, h1, ntiles);  // 8 waves = 8 heads / tile
  k_alpha1<<<(N * NHEADSC + 255) / 256, 256, 0, stream>>>(h1, a_att, asc1, ads1, N);
  k_edge1<<<(E + 7) / 8, 256, 0, stream>>>(row, col, asc1, ads1, h1, hp1, rs1, E);
  k_norm1<<<(int)((NS + 255) / 256), 256, 0, stream>>>(hp1, rs1, (int)NS);

  // ---- layer 2 ----
  hipMemsetAsync(hp2, 0, (size_t)N * 64 * sizeof(float), stream);
  hipMemsetAsync(rs2, 0, (size_t)N * sizeof(float), stream);

  k_gemm2<<<(ntiles * 4 + 7) / 8, 256, 0, stream>>>(hp1, Wout, h2, ntiles);
  k_alpha2<<<(N + 255) / 256, 256, 0, stream>>>(h2, a_out, asc2, ads2, N);
  k_edge2<<<(E + 7) / 8, 256, 0, stream>>>(row, col, asc2, ads2, h2, hp2, rs2, E);
  k_norm2<<<(N * 64 + 255) / 256, 256, 0, stream>>>(hp2, rs2, xov, N * 64);

  // ---- MLP head ----
  k_mlp<<<ntiles, 32, 0, stream>>>(xov, W1, b1, W2, b2, W3, b3, out, ntiles);
}